// FISLayerParameterSharingV2_584115552380
// MI455X (gfx1250) — compile-verified
//
#include <hip/hip_runtime.h>

#define NB 32    // batch
#define NT 32    // trees
#define NC 64    // channels
#define NH 128
#define NW 128
#define NHW (NH*NW)

#define XPITCH 66   // padded LDS pitch for x tile  [w][c], conflict-free, 8B-aligned
#define APITCH 66   // padded LDS pitch for alpha tile [m][c]
#define MMAX   96   // max stacked-A rows (3*T)

typedef float v2f __attribute__((ext_vector_type(2)));
typedef float v8f __attribute__((ext_vector_type(8)));

// ---------------------------------------------------------------------------
// Kernel 1: stacked-A GEMM via V_WMMA_F32_16X16X4_F32.
//   Z row m = k*tchunk + tt  holds  sum_c alpha_k[t0+tt, c] * x[b, c, h, :]
// FULL=1: Z layout [m][b][h][w], m padded to mtiles*16 -> unguarded stores
//         with constant per-r strides (no per-lane div/mod, no branches).
// FULL=0: compact Z layout [k][b][tt][h][w] with guarded stores (small ws).
// Both operands staged in LDS; B fragments register-resident across M-tiles.
// ---------------------------------------------------------------------------
template<bool FULL>
__global__ __launch_bounds__(256) void zgemm_wmma(
    const float* __restrict__ x,
    const float* __restrict__ a1,
    const float* __restrict__ a2,
    const float* __restrict__ a3,
    float* __restrict__ z,
    int t0, int tchunk, int ltc, int mtiles)
{
    __shared__ __align__(16) float xs[NW * XPITCH];   // [w][c]  (~33.8 KB)
    __shared__ __align__(16) float as[MMAX * APITCH]; // [m][c]  (~25.3 KB)

    const int b = blockIdx.x / NH;
    const int h = blockIdx.x % NH;
    const int mrows = 3 * tchunk;
    const int mpad  = mtiles * 16;

    // ---- stage x[b, :, h, :] transposed into LDS: xs[w][c] ----------------
    const float* xrow = x + (size_t)b * NC * NHW + (size_t)h * NW;
    for (int i = threadIdx.x; i < (NC * NW) / 4; i += blockDim.x) {
        const int c  = i >> 5;          // 32 float4 per channel row
        const int w4 = (i & 31) * 4;
        const float4 v = *((const float4*)(xrow + (size_t)c * NHW + w4));
        xs[(w4 + 0) * XPITCH + c] = v.x;
        xs[(w4 + 1) * XPITCH + c] = v.y;
        xs[(w4 + 2) * XPITCH + c] = v.z;
        xs[(w4 + 3) * XPITCH + c] = v.w;
    }

    // ---- stage stacked alpha tile (zero-padded) into LDS: as[m][c] --------
    for (int i = threadIdx.x; i < mpad * NC; i += blockDim.x) {
        const int m = i >> 6;           // NC = 64
        const int c = i & 63;
        float v = 0.0f;
        if (m < mrows) {
            const int k  = m >> ltc;
            const int tt = m & (tchunk - 1);
            const float* am = (k == 0) ? a1 : ((k == 1) ? a2 : a3);
            v = am[(size_t)(t0 + tt) * NC + c];
        }
        as[m * APITCH + c] = v;
    }
    __syncthreads();

    const int lane = threadIdx.x & 31;
    const int wave = threadIdx.x >> 5;     // 8 waves -> 8 N-tiles of 16 pixels
    const int lm   = lane & 15;
    const int hs   = lane >> 4;            // half-select: K += 2 for lanes 16-31
    const int w0   = wave * 16;

    // ---- B fragments: invariant across M-tiles, keep in registers --------
    v2f bf[16];
    const float* bbase = xs + (w0 + lm) * XPITCH + 2 * hs;
    #pragma unroll
    for (int ks = 0; ks < 16; ++ks)
        bf[ks] = *(const v2f*)(bbase + ks * 4);

    for (int mt = 0; mt < mtiles; ++mt) {
        const float* abase = as + (mt * 16 + lm) * APITCH + 2 * hs;

        v2f af[16];
        #pragma unroll
        for (int ks = 0; ks < 16; ++ks)
            af[ks] = *(const v2f*)(abase + ks * 4);

        v8f acc = {};
        #pragma unroll
        for (int ks = 0; ks < 16; ++ks)
            acc = __builtin_amdgcn_wmma_f32_16x16x4_f32(
                false, af[ks], false, bf[ks], (short)0, acc, false, false);

        // D 16x16 f32: VGPR r -> M = mt*16 + r + 8*hs, N = lm
        if (FULL) {
            float* zb = z + ((size_t)(mt * 16 + 8 * hs) * NB + b) * NHW
                          + (size_t)h * NW + w0 + lm;
            #pragma unroll
            for (int r = 0; r < 8; ++r)
                zb[(size_t)r * (NB * NHW)] = acc[r];
        } else {
            #pragma unroll
            for (int r = 0; r < 8; ++r) {
                const int ml = mt * 16 + r + 8 * hs;
                if (ml < mrows) {
                    const int kz = ml >> ltc;
                    const int tz = ml & (tchunk - 1);
                    z[(((size_t)kz * NB + b) * tchunk + tz) * NHW
                      + (size_t)h * NW + w0 + lm] = acc[r];
                }
            }
        }
    }
}

// ---------------------------------------------------------------------------
// Discounted inclusive scan across 128 threads (4 waves of 32):
// y[i] = sum_{j<=i} d^{i-j} x[j].
// Intra-wave: Hillis-Steele with weights d^{2^k} via __shfl_up (wave32).
// Inter-wave: carry = sum_{V<W} d^{32(W-V-1)} * total_V, applied as d^{L+1}*carry.
// ---------------------------------------------------------------------------
__device__ __forceinline__ float dscan128(float v, float d, float dlp1, float d32,
                                          float* tot, int lane, int wave)
{
    float dp = d;
    #pragma unroll
    for (int off = 1; off < 32; off <<= 1) {
        float o = __shfl_up(v, off, 32);
        if (lane >= off) v = fmaf(dp, o, v);
        dp *= dp;                       // d^1, d^2, d^4, d^8, d^16
    }
    if (lane == 31) tot[wave] = v;
    __syncthreads();
    float c = 0.0f, s = 1.0f;
    for (int j = wave - 1; j >= 0; --j) { c = fmaf(s, tot[j], c); s *= d32; }
    v = fmaf(dlp1, c, v);
    __syncthreads();                    // tot[] safe to reuse
    return v;
}

// ---------------------------------------------------------------------------
// Kernel 2: fused causal pipeline per (b,t):
//   A3 = dcs2(z3); s3 = d^2 * A3[h-1,w-1]
//   A2 = dcs2(z2*s3); s2 = d^2 * A2[h-1,w-1]
//   out = dcs2(z1*s2)
// Thread = column w; raster order over rows; H-scan state in registers,
// W-scan via dscan128; previous-row shifted values via small LDS rows.
// ---------------------------------------------------------------------------
__global__ __launch_bounds__(128) void scan_chain(
    const float* __restrict__ z,
    const float* __restrict__ disc,
    float* __restrict__ out,
    int t0, int tchunk, int full)
{
    __shared__ float a3prev[NW];
    __shared__ float a2prev[NW];
    __shared__ float tot[4];

    const int bt = blockIdx.x;
    const int b  = bt / tchunk;
    const int tt = bt % tchunk;
    const int t  = t0 + tt;
    const int w    = threadIdx.x;
    const int lane = w & 31;
    const int wave = w >> 5;

    const float d    = disc[t];
    const float d2   = d * d;
    const float dlp1 = __powf(d, (float)(lane + 1));
    const float d32  = __powf(d, 32.0f);

    // Z row base for k = 0,1,2 in the active layout
    size_t o1, o2, o3;
    if (full) {                         // [m][b][h][w], m = k*tchunk + tt
        o1 = ((size_t)(0 * tchunk + tt) * NB + b) * NHW;
        o2 = ((size_t)(1 * tchunk + tt) * NB + b) * NHW;
        o3 = ((size_t)(2 * tchunk + tt) * NB + b) * NHW;
    } else {                            // [k][b][tt][h][w]
        o1 = (((size_t)0 * NB + b) * tchunk + tt) * NHW;
        o2 = (((size_t)1 * NB + b) * tchunk + tt) * NHW;
        o3 = (((size_t)2 * NB + b) * tchunk + tt) * NHW;
    }
    const float* z1p = z + o1;
    const float* z2p = z + o2;
    const float* z3p = z + o3;
    float* outp = out + ((size_t)b * NT + t) * NHW;

    float c3 = 0.0f, c2 = 0.0f, c1 = 0.0f;   // H-scan (column) states
    a3prev[w] = 0.0f;
    a2prev[w] = 0.0f;
    __syncthreads();

    for (int h = 0; h < NH; ++h) {
        const float s3 = (w > 0) ? d2 * a3prev[w - 1] : 0.0f;
        const float s2 = (w > 0) ? d2 * a2prev[w - 1] : 0.0f;

        const float z1v = z1p[h * NW + w];
        const float z2v = z2p[h * NW + w];
        const float z3v = z3p[h * NW + w];
        if (h + 1 < NH) {
            __builtin_prefetch(z1p + (h + 1) * NW + w, 0, 0);
            __builtin_prefetch(z2p + (h + 1) * NW + w, 0, 0);
            __builtin_prefetch(z3p + (h + 1) * NW + w, 0, 0);
        }

        c3 = fmaf(d, c3, z3v);
        const float A3 = dscan128(c3, d, dlp1, d32, tot, lane, wave);

        c2 = fmaf(d, c2, z2v * s3);
        const float A2 = dscan128(c2, d, dlp1, d32, tot, lane, wave);

        c1 = fmaf(d, c1, z1v * s2);
        const float A1 = dscan128(c1, d, dlp1, d32, tot, lane, wave);

        outp[h * NW + w] = A1;

        __syncthreads();          // all reads of a*prev (row h) complete
        a3prev[w] = A3;
        a2prev[w] = A2;
        __syncthreads();          // writes visible for row h+1
    }
}

// ---------------------------------------------------------------------------
// Host launcher: prefer padded M-major layout (branch-free GEMM stores),
// fall back to compact layout if the workspace is small. Chunk T to fit ws.
// ---------------------------------------------------------------------------
extern "C" void kernel_launch(void* const* d_in, const int* in_sizes, int n_in,
                              void* d_out, int out_size, void* d_ws, size_t ws_size,
                              hipStream_t stream)
{
    const float* x  = (const float*)d_in[0];
    const float* a1 = (const float*)d_in[1];
    const float* a2 = (const float*)d_in[2];
    const float* a3 = (const float*)d_in[3];
    const float* dc = (const float*)d_in[4];
    float* out = (float*)d_out;
    float* z   = (float*)d_ws;

    int tchunk = 1, mtiles = 1, full = 0;
    for (int tc = NT; tc >= 1; tc >>= 1) {          // padded layout first
        const int mt = (3 * tc + 15) / 16;
        if ((size_t)mt * 16 * NB * NHW * sizeof(float) <= ws_size) {
            tchunk = tc; mtiles = mt; full = 1; break;
        }
    }
    if (!full) {                                     // compact fallback
        tchunk = NT;
        while (tchunk > 1 &&
               (size_t)3 * NB * (size_t)tchunk * NHW * sizeof(float) > ws_size)
            tchunk >>= 1;
        mtiles = (3 * tchunk + 15) / 16;
    }
    int ltc = 0;
    while ((1 << (ltc + 1)) <= tchunk) ++ltc;        // log2(tchunk)

    for (int t0 = 0; t0 < NT; t0 += tchunk) {
        if (full)
            zgemm_wmma<true><<<NB * NH, 256, 0, stream>>>(
                x, a1, a2, a3, z, t0, tchunk, ltc, mtiles);
        else
            zgemm_wmma<false><<<NB * NH, 256, 0, stream>>>(
                x, a1, a2, a3, z, t0, tchunk, ltc, mtiles);
        scan_chain<<<NB * tchunk, 128, 0, stream>>>(z, dc, out, t0, tchunk, full);
    }
}